// GRUObservationCellLogvar_63273458205108
// MI455X (gfx1250) — compile-verified
//
#include <hip/hip_runtime.h>
#include <math.h>

typedef float v2f __attribute__((ext_vector_type(2)));
typedef float v4f __attribute__((ext_vector_type(4)));
typedef float v8f __attribute__((ext_vector_type(8)));

#define DCH   32
#define HID   128
#define PREP  4
#define ROWS  48     // observations per block = three 16-row WMMA M tiles
#define RT    3      // row tiles
#define ASTR  132    // padded LDS row stride (floats) -> conflict-free ds_load_b64
#define PSTR  388    // padded gate-preact row stride

__device__ __forceinline__ float fast_sigmoid(float x) {
    return 1.f / (1.f + __expf(-x));
}
__device__ __forceinline__ float fast_tanh(float x) {
    float ax = fabsf(x);
    float e  = __expf(-2.f * ax);
    float t  = (1.f - e) / (1.f + e);
    return copysignf(t, x);
}

// ---------- helper kernels: flagged bulk copy of unobserved rows -------------
__global__ void zero_flags_kernel(uint4* __restrict__ f, int n16) {
    int g = blockIdx.x * 256 + threadIdx.x;
    if (g < n16) f[g] = make_uint4(0u, 0u, 0u, 0u);
}

__global__ void set_flags_kernel(unsigned char* __restrict__ f,
                                 const int* __restrict__ i_obs, int n_obs) {
    int g = blockIdx.x * 256 + threadIdx.x;
    if (g < n_obs) f[i_obs[g]] = 1;
}

// one wave per row (32 lanes x float4 = 128 floats); flag read is wave-uniform
__global__ void copy_unobserved_kernel(const float* __restrict__ h,
                                       float* __restrict__ out,
                                       const unsigned char* __restrict__ f,
                                       int nrows) {
    int g   = blockIdx.x * 256 + threadIdx.x;
    int row = g >> 5;
    int c4  = (g & 31) * 4;
    if (row < nrows && f[row] == 0) {
        v4f v = __builtin_nontemporal_load(
                    reinterpret_cast<const v4f*>(&h[(size_t)row * HID + c4]));
        __builtin_nontemporal_store(
            v, reinterpret_cast<v4f*>(&out[(size_t)row * HID + c4]));
    }
}

// ---------------------------- main fused kernel ------------------------------
__launch_bounds__(256, 2)
__global__ void gru_obs_wmma_kernel(
    const float* __restrict__ h,
    const float* __restrict__ p,
    const float* __restrict__ X_obs,
    const float* __restrict__ M_obs,
    const int*   __restrict__ i_obs,
    const float* __restrict__ w_prep,
    const float* __restrict__ bias_prep,
    const float* __restrict__ W_ih,
    const float* __restrict__ W_hh,
    const float* __restrict__ b_ih,
    const float* __restrict__ b_hh,
    float* __restrict__ out_h,
    float* __restrict__ out_losses,
    int n_obs)
{
    __shared__ __align__(16) float Agi[ROWS * ASTR];  // gru_in tiles [48 x 128]
    __shared__ __align__(16) float Ah [ROWS * ASTR];  // h_obs tiles  [48 x 128]
    __shared__ __align__(16) float Pre[ROWS * PSTR];  // gate preact  [48 x 384]
    __shared__ __align__(16) float Hn [ROWS * ASTR];  // h_n part     [48 x 128]
    __shared__ int Idx[ROWS];

    const int t      = threadIdx.x;
    const int lane   = t & 31;
    const int w      = t >> 5;            // wave id 0..7
    const int block0 = blockIdx.x * ROWS;

    if (t < ROWS) {
        int og = block0 + t;
        Idx[t] = (og < n_obs) ? i_obs[og] : -1;
    }
    __syncthreads();

    // ---------------- Stage 1: losses + prep net -> A matrices in LDS --------
    #pragma unroll
    for (int it = 0; it < (ROWS * DCH) / 256; ++it) {
        int pair = t + it * 256;          // (obs row, channel)
        int o = pair >> 5;
        int d = pair & 31;
        int og  = block0 + o;
        int idx = Idx[o];
        if (idx >= 0) {
            float mean   = p[(size_t)idx * (2 * DCH) + d];
            float logvar = p[(size_t)idx * (2 * DCH) + DCH + d];
            float xv     = X_obs[(size_t)og * DCH + d];
            float m      = M_obs[(size_t)og * DCH + d];
            float lv     = fminf(fmaxf(logvar, -10.f), 10.f);
            float sigma  = fminf(fmaxf(__expf(0.5f * lv), 1e-6f), 1e6f);
            float err    = fminf(fmaxf((xv - mean) / sigma, -1e6f), 1e6f);
            out_losses[(size_t)og * DCH + d] =
                0.5f * (err * err + lv + 1.8378770664093453f) * m;
            float f0 = xv, f1 = mean, f2 = lv, f3 = err;
            #pragma unroll
            for (int pp = 0; pp < PREP; ++pp) {
                float g = bias_prep[d * PREP + pp]
                        + f0 * w_prep[d * 16 + 0 * PREP + pp]
                        + f1 * w_prep[d * 16 + 1 * PREP + pp]
                        + f2 * w_prep[d * 16 + 2 * PREP + pp]
                        + f3 * w_prep[d * 16 + 3 * PREP + pp];
                Agi[o * ASTR + d * PREP + pp] = fmaxf(g, 0.f) * m;
            }
        } else {
            #pragma unroll
            for (int pp = 0; pp < PREP; ++pp)
                Agi[o * ASTR + d * PREP + pp] = 0.f;
        }
    }
    // gather h_obs tiles: ROWS*128 floats, float4 per thread per iter
    #pragma unroll
    for (int i = 0; i < (ROWS * HID / 4) / 256; ++i) {
        int lin = t + i * 256;            // (row, 16B group)
        int o  = lin >> 5;                // 32 groups per row
        int c4 = (lin & 31) * 4;
        int idx = Idx[o];
        float4 v = make_float4(0.f, 0.f, 0.f, 0.f);
        if (idx >= 0)
            v = *reinterpret_cast<const float4*>(&h[(size_t)idx * HID + c4]);
        *reinterpret_cast<float4*>(&Ah[o * ASTR + c4]) = v;
    }
    __syncthreads();

    // ---------------- Stage 2: fp32 WMMA GEMMs -------------------------------
    // A frag (16x4 f32): lane L -> row M = L&15, K = 4*ks + 2*(L>>4) + {0,1}
    // B frag (4x16 f32): lane L -> col N = L&15, same K split
    const int mrow  = lane & 15;
    const int khalf = lane >> 4;

    // r and z gate tiles: both GEMMs sum into one accumulator per row tile
    #pragma unroll
    for (int g = 0; g < 2; ++g) {
        const int tile = w + 8 * g;
        const int n    = tile * 16 + mrow;          // gate column 0..255
        const float binit = b_ih[n] + b_hh[n];
        v8f acc[RT];
        #pragma unroll
        for (int rt = 0; rt < RT; ++rt)
            #pragma unroll
            for (int r = 0; r < 8; ++r) acc[rt][r] = binit;
        #pragma unroll
        for (int ks = 0; ks < 32; ++ks) {
            const int k = ks * 4 + 2 * khalf;
            v2f b = *reinterpret_cast<const v2f*>(&W_ih[n * HID + k]);
            #pragma unroll
            for (int rt = 0; rt < RT; ++rt) {
                v2f a = *reinterpret_cast<const v2f*>(&Agi[(rt * 16 + mrow) * ASTR + k]);
                acc[rt] = __builtin_amdgcn_wmma_f32_16x16x4_f32(
                              false, a, false, b, (short)0, acc[rt], false, false);
            }
        }
        #pragma unroll
        for (int ks = 0; ks < 32; ++ks) {
            const int k = ks * 4 + 2 * khalf;
            v2f b = *reinterpret_cast<const v2f*>(&W_hh[n * HID + k]);
            #pragma unroll
            for (int rt = 0; rt < RT; ++rt) {
                v2f a = *reinterpret_cast<const v2f*>(&Ah[(rt * 16 + mrow) * ASTR + k]);
                acc[rt] = __builtin_amdgcn_wmma_f32_16x16x4_f32(
                              false, a, false, b, (short)0, acc[rt], false, false);
            }
        }
        #pragma unroll
        for (int rt = 0; rt < RT; ++rt)
            #pragma unroll
            for (int r = 0; r < 8; ++r)
                Pre[(rt * 16 + r + 8 * khalf) * PSTR + n] = acc[rt][r];
    }
    // n-gate tile: i_n and h_n must stay separate (n = tanh(i_n + r*h_n))
    {
        const int n = (16 + w) * 16 + mrow;          // gate column 256..383
        const float bi = b_ih[n], bh = b_hh[n];
        v8f accI[RT], accH[RT];
        #pragma unroll
        for (int rt = 0; rt < RT; ++rt)
            #pragma unroll
            for (int r = 0; r < 8; ++r) { accI[rt][r] = bi; accH[rt][r] = bh; }
        #pragma unroll
        for (int ks = 0; ks < 32; ++ks) {
            const int k = ks * 4 + 2 * khalf;
            v2f b = *reinterpret_cast<const v2f*>(&W_ih[n * HID + k]);
            #pragma unroll
            for (int rt = 0; rt < RT; ++rt) {
                v2f a = *reinterpret_cast<const v2f*>(&Agi[(rt * 16 + mrow) * ASTR + k]);
                accI[rt] = __builtin_amdgcn_wmma_f32_16x16x4_f32(
                               false, a, false, b, (short)0, accI[rt], false, false);
            }
        }
        #pragma unroll
        for (int ks = 0; ks < 32; ++ks) {
            const int k = ks * 4 + 2 * khalf;
            v2f b = *reinterpret_cast<const v2f*>(&W_hh[n * HID + k]);
            #pragma unroll
            for (int rt = 0; rt < RT; ++rt) {
                v2f a = *reinterpret_cast<const v2f*>(&Ah[(rt * 16 + mrow) * ASTR + k]);
                accH[rt] = __builtin_amdgcn_wmma_f32_16x16x4_f32(
                               false, a, false, b, (short)0, accH[rt], false, false);
            }
        }
        #pragma unroll
        for (int rt = 0; rt < RT; ++rt)
            #pragma unroll
            for (int r = 0; r < 8; ++r) {
                const int row = rt * 16 + r + 8 * khalf;
                Pre[row * PSTR + n]         = accI[rt][r];
                Hn [row * ASTR + (n - 256)] = accH[rt][r];
            }
    }
    __syncthreads();

    // ---------------- Stage 3: gate combine + scatter (float4 wide) ----------
    #pragma unroll
    for (int i = 0; i < (ROWS * HID / 4) / 256; ++i) {
        int lin = t + i * 256;
        int o  = lin >> 5;
        int c  = (lin & 31) * 4;
        int idx = Idx[o];
        if (idx >= 0) {
            float4 rp = *reinterpret_cast<const float4*>(&Pre[o * PSTR + c]);
            float4 zp = *reinterpret_cast<const float4*>(&Pre[o * PSTR + 128 + c]);
            float4 ip = *reinterpret_cast<const float4*>(&Pre[o * PSTR + 256 + c]);
            float4 hn = *reinterpret_cast<const float4*>(&Hn [o * ASTR + c]);
            float4 hp = *reinterpret_cast<const float4*>(&Ah [o * ASTR + c]);
            float4 out;
            {
                float rg = fast_sigmoid(rp.x), zg = fast_sigmoid(zp.x);
                float ng = fast_tanh(ip.x + rg * hn.x);
                out.x = (1.f - zg) * ng + zg * hp.x;
            }
            {
                float rg = fast_sigmoid(rp.y), zg = fast_sigmoid(zp.y);
                float ng = fast_tanh(ip.y + rg * hn.y);
                out.y = (1.f - zg) * ng + zg * hp.y;
            }
            {
                float rg = fast_sigmoid(rp.z), zg = fast_sigmoid(zp.z);
                float ng = fast_tanh(ip.z + rg * hn.z);
                out.z = (1.f - zg) * ng + zg * hp.z;
            }
            {
                float rg = fast_sigmoid(rp.w), zg = fast_sigmoid(zp.w);
                float ng = fast_tanh(ip.w + rg * hn.w);
                out.w = (1.f - zg) * ng + zg * hp.w;
            }
            *reinterpret_cast<float4*>(&out_h[(size_t)idx * HID + c]) = out;
        }
    }
}

extern "C" void kernel_launch(void* const* d_in, const int* in_sizes, int n_in,
                              void* d_out, int out_size, void* d_ws, size_t ws_size,
                              hipStream_t stream) {
    const float* h         = (const float*)d_in[0];
    const float* p         = (const float*)d_in[1];
    const float* X_obs     = (const float*)d_in[2];
    const float* M_obs     = (const float*)d_in[3];
    const int*   i_obs     = (const int*)  d_in[4];
    const float* w_prep    = (const float*)d_in[5];
    const float* bias_prep = (const float*)d_in[6];
    const float* W_ih      = (const float*)d_in[7];
    const float* W_hh      = (const float*)d_in[8];
    const float* b_ih      = (const float*)d_in[9];
    const float* b_hh      = (const float*)d_in[10];

    const int N     = in_sizes[0] / HID;    // 500000
    const int n_obs = in_sizes[4];          // 200000

    float* out_h      = (float*)d_out;
    float* out_losses = out_h + (size_t)N * HID;

    // Bulk h -> out_h stream. If workspace allows, skip rows that the GRU
    // kernel will overwrite (saves ~200 MB of HBM traffic); NT hints keep the
    // weight matrices resident in L2.
    if (ws_size >= (size_t)N) {
        unsigned char* flags = (unsigned char*)d_ws;
        int n16 = (N + 15) / 16;
        zero_flags_kernel<<<(n16 + 255) / 256, 256, 0, stream>>>(
            (uint4*)d_ws, n16);
        set_flags_kernel<<<(n_obs + 255) / 256, 256, 0, stream>>>(
            flags, i_obs, n_obs);
        long total = (long)N * 32;
        copy_unobserved_kernel<<<(int)((total + 255) / 256), 256, 0, stream>>>(
            h, out_h, flags, N);
    } else {
        hipMemcpyAsync(out_h, h, (size_t)N * HID * sizeof(float),
                       hipMemcpyDeviceToDevice, stream);
    }

    const int blocks = (n_obs + ROWS - 1) / ROWS;   // 4167
    gru_obs_wmma_kernel<<<blocks, 256, 0, stream>>>(
        h, p, X_obs, M_obs, i_obs, w_prep, bias_prep,
        W_ih, W_hh, b_ih, b_hh, out_h, out_losses, n_obs);
}